// TinyCrossMamba_78658031059013
// MI455X (gfx1250) — compile-verified
//
#include <hip/hip_runtime.h>
#include <hip/hip_bf16.h>
#include <math.h>

// ---------------- model constants ----------------
#define D_MODEL 768
#define D_INNER 1536
#define D_STATE 16
#define D_CONV  4
#define DT_RANK 48
#define N_LAYER 4
#define VOCAB   32000
#define BATCH   2
#define SEQLEN  1024
#define ROWS    (BATCH * SEQLEN)        // 2048 token rows

typedef __attribute__((ext_vector_type(2))) float v2f;
typedef __attribute__((ext_vector_type(8))) float v8f;

__device__ __forceinline__ float silu_f(float x) {
    return x / (1.0f + __expf(-x));
}

// =====================================================================
// LDS-staged fp32 WMMA GEMM with CDNA5 async global->LDS copy.
//   Y[m,n] = act( sum_k X[m,k]*W[n,k] + bias[n] ) (+ resid[m,n])
// Block = 256 threads = 8 wave32. Tile: BM=128 rows (one 16-row M-tile per
// wave) x BN=64 cols (4 N-tiles per wave). The 64xBK B-strip is shared by all
// 8 waves and is double-buffered in LDS via global_load_async_to_lds_b128
// (ASYNCcnt), with LDS row stride 20 floats -> bank-conflict-free ds_load_b64
// fragments and 16B-aligned B128 fills.
// Requires: M % 128 == 0, N % 64 == 0, K % 16 == 0.
// act: 0=none, 1=silu, 2=softplus
// =====================================================================
#define GBM 128
#define GBN 64
#define GBK 16
#define BROW 20   // padded LDS row stride (floats)

__global__ __launch_bounds__(256)
void gemm_wmma_f32_lds(const float* __restrict__ X, const float* __restrict__ W,
                       float* __restrict__ Y,
                       const float* __restrict__ bias, const float* __restrict__ resid,
                       int M, int N, int K, int ldx, int ldw, int ldy, int act)
{
    __shared__ float sB[2][GBN * BROW];

    const int lane   = threadIdx.x & 31;
    const int wave   = threadIdx.x >> 5;
    const int tilesN = N / GBN;
    const int mbase  = (blockIdx.x / tilesN) * GBM;
    const int nbase  = (blockIdx.x % tilesN) * GBN;

    // cooperative fill mapping: 256 threads x 1 float4 = 64 rows x 16 floats
    const int frow = threadIdx.x >> 2;          // 0..63
    const int fcol = (threadIdx.x & 3) * 4;     // 0,4,8,12
    const float* gfill = W + (size_t)(nbase + frow) * ldw + fcol;
    const unsigned lds0 = (unsigned)(uintptr_t)(&sB[0][frow * BROW + fcol]);
    const unsigned lds1 = (unsigned)(uintptr_t)(&sB[1][frow * BROW + fcol]);

    // A-fragment addressing (fp32 16x4 layout: lanes 16-31 mirror rows, K+2)
    const int row  = mbase + wave * 16 + (lane & 15);
    const int koff = (lane >> 4) * 2;
    const float* xrow = X + (size_t)row * ldx + koff;

    v8f acc[4];
#pragma unroll
    for (int j = 0; j < 4; ++j) acc[j] = (v8f){0,0,0,0,0,0,0,0};

    // prologue: async-fill chunk 0 into buffer 0
    asm volatile("global_load_async_to_lds_b128 %0, %1, off"
                 :: "v"(lds0), "v"(gfill) : "memory");

    for (int kc = 0; kc < K; kc += GBK) {
        const int cur = (kc / GBK) & 1;
        asm volatile("s_wait_asynccnt 0x0" ::: "memory");
        __syncthreads();                          // chunk visible to all waves
        if (kc + GBK < K) {                       // overlap next fill with compute
            const float* gnext = gfill + kc + GBK;
            const unsigned ldsn = cur ? lds0 : lds1;
            asm volatile("global_load_async_to_lds_b128 %0, %1, off"
                         :: "v"(ldsn), "v"(gnext) : "memory");
        }
        const float* sbuf = &sB[cur][0];
#pragma unroll
        for (int kk = 0; kk < GBK; kk += 4) {
            v2f a = *(const v2f*)(xrow + kc + kk);
#pragma unroll
            for (int j = 0; j < 4; ++j) {
                v2f b = *(const v2f*)(sbuf + (j * 16 + (lane & 15)) * BROW + kk + koff);
                acc[j] = __builtin_amdgcn_wmma_f32_16x16x4_f32(
                            false, a, false, b, (short)0, acc[j], false, false);
            }
        }
        // next iteration's wait+barrier orders buffer reuse; no extra barrier
    }

    // epilogue (C/D 16x16 f32 layout)
    const int mwr   = mbase + wave * 16 + (lane >> 4) * 8;
    const int ncol0 = nbase + (lane & 15);
#pragma unroll
    for (int j = 0; j < 4; ++j) {
        const int n  = ncol0 + j * 16;
        const float bb = bias ? bias[n] : 0.0f;
#pragma unroll
        for (int r = 0; r < 8; ++r) {
            const int m = mwr + r;
            float v = acc[j][r] + bb;
            if (act == 1)      v = silu_f(v);
            else if (act == 2) v = (v > 20.0f) ? v : log1pf(__expf(v));
            size_t idx = (size_t)m * ldy + n;
            if (resid) v += resid[idx];
            Y[idx] = v;
        }
    }
}

// =====================================================================
// Direct WMMA GEMM (no LDS) for small-N cases (c_proj, N=16).
// =====================================================================
template<int NT>
__global__ __launch_bounds__(256)
void gemm_wmma_f32(const float* __restrict__ X, const float* __restrict__ W,
                   float* __restrict__ Y,
                   const float* __restrict__ bias, const float* __restrict__ resid,
                   int M, int N, int K, int ldx, int ldw, int ldy, int act)
{
    const int lane   = threadIdx.x & 31;
    const int waveId = (blockIdx.x * (blockDim.x >> 5)) + (threadIdx.x >> 5);
    const int tilesN = N / (16 * NT);
    const int tm     = waveId / tilesN;
    const int tn     = waveId % tilesN;
    if (tm * 16 >= M) return;

    const int row  = tm * 16 + (lane & 15);
    const int koff = (lane >> 4) * 2;
    const int nb   = tn * (16 * NT) + (lane & 15);

    v8f acc[NT];
#pragma unroll
    for (int j = 0; j < NT; ++j) acc[j] = (v8f){0,0,0,0,0,0,0,0};

    const float* xrow = X + (size_t)row * ldx + koff;

    for (int k = 0; k < K; k += 4) {
        v2f a = *(const v2f*)(xrow + k);
#pragma unroll
        for (int j = 0; j < NT; ++j) {
            const float* wrow = W + (size_t)(nb + j * 16) * ldw + koff;
            v2f b = *(const v2f*)(wrow + k);
            acc[j] = __builtin_amdgcn_wmma_f32_16x16x4_f32(
                        false, a, false, b, (short)0, acc[j], false, false);
        }
    }

    const int mbase = tm * 16 + (lane >> 4) * 8;
    const int ncol0 = tn * (16 * NT) + (lane & 15);
#pragma unroll
    for (int j = 0; j < NT; ++j) {
        const int n  = ncol0 + j * 16;
        const float bb = bias ? bias[n] : 0.0f;
#pragma unroll
        for (int r = 0; r < 8; ++r) {
            const int m = mbase + r;
            float v = acc[j][r] + bb;
            if (act == 1)      v = silu_f(v);
            else if (act == 2) v = (v > 20.0f) ? v : log1pf(__expf(v));
            size_t idx = (size_t)m * ldy + n;
            if (resid) v += resid[idx];
            Y[idx] = v;
        }
    }
}

// ---------------- embedding gather ----------------
__global__ __launch_bounds__(256)
void embed_kernel(const int* __restrict__ tok, const float* __restrict__ emb,
                  float* __restrict__ x, int total)
{
    int i = blockIdx.x * blockDim.x + threadIdx.x;
    if (i >= total) return;
    int r = i / D_MODEL, c = i % D_MODEL;
    x[i] = emb[(size_t)tok[r] * D_MODEL + c];
}

// ---------------- RMSNorm (one block per token row) ----------------
__global__ __launch_bounds__(256)
void rmsnorm_kernel(const float* __restrict__ x, const float* __restrict__ w,
                    float* __restrict__ out, int D)
{
    __shared__ float red[9];
    const int row = blockIdx.x;
    const float* xr = x + (size_t)row * D;
    float s = 0.0f;
    for (int c = threadIdx.x; c < D; c += blockDim.x) { float v = xr[c]; s += v * v; }
    for (int m = 16; m >= 1; m >>= 1) s += __shfl_xor(s, m);
    const int wave = threadIdx.x >> 5, lane = threadIdx.x & 31;
    if (lane == 0) red[wave] = s;
    __syncthreads();
    if (wave == 0) {
        s = (lane < (int)(blockDim.x >> 5)) ? red[lane] : 0.0f;
        for (int m = 4; m >= 1; m >>= 1) s += __shfl_xor(s, m);
        if (lane == 0) red[8] = rsqrtf(s / (float)D + 1e-5f);
    }
    __syncthreads();
    const float scale = red[8];
    for (int c = threadIdx.x; c < D; c += blockDim.x)
        out[(size_t)row * D + c] = xr[c] * scale * w[c];
}

// ---------------- depthwise causal conv + bias + SiLU ----------------
__global__ __launch_bounds__(256)
void conv_silu_kernel(const float* __restrict__ xr, const float* __restrict__ w,
                      const float* __restrict__ bias, float* __restrict__ t, int total)
{
    int i = blockIdx.x * blockDim.x + threadIdx.x;
    if (i >= total) return;
    const int d = i % D_INNER;
    const int l = (i / D_INNER) % SEQLEN;
    const int b = i / (D_INNER * SEQLEN);
    float acc = bias[d];
#pragma unroll
    for (int k = 0; k < D_CONV; ++k) {
        const int ls = l - (D_CONV - 1) + k;
        if (ls >= 0)
            acc += xr[((size_t)(b * SEQLEN + ls)) * (2 * D_INNER) + d] * w[d * D_CONV + k];
    }
    t[i] = silu_f(acc);
}

// ---------------- selective scan ----------------
__global__ __launch_bounds__(256)
void scan_kernel(const float* __restrict__ u, const float* __restrict__ delta,
                 const float* __restrict__ xdbl,   // stride DT_RANK+D_STATE, B at +DT_RANK
                 const float* __restrict__ Cbuf,   // stride D_STATE
                 const float* __restrict__ A_log, const float* __restrict__ Dp,
                 float* __restrict__ y)
{
    const int lane = threadIdx.x & 31;
    const int n    = lane & 15;
    const int dsub = lane >> 4;
    const int wave = threadIdx.x >> 5;
    const int dPerBlock     = (blockDim.x >> 5) * 2;         // 16
    const int blocksPerBat  = D_INNER / dPerBlock;           // 96
    const int b  = blockIdx.x / blocksPerBat;
    const int d  = (blockIdx.x % blocksPerBat) * dPerBlock + wave * 2 + dsub;

    const float A  = -__expf(A_log[d * D_STATE + n]);
    const float Dd = Dp[d];
    float h = 0.0f;

    for (int l = 0; l < SEQLEN; ++l) {
        const size_t r = (size_t)b * SEQLEN + l;
        const float uu = u[r * D_INNER + d];
        const float dl = delta[r * D_INNER + d];
        const float Bv = xdbl[r * (DT_RANK + D_STATE) + DT_RANK + n];
        const float Cv = Cbuf[r * D_STATE + n];
        h = __expf(dl * A) * h + dl * Bv * uu;
        float p = h * Cv;
        p += __shfl_xor(p, 1);
        p += __shfl_xor(p, 2);
        p += __shfl_xor(p, 4);
        p += __shfl_xor(p, 8);
        if (n == 0) y[r * D_INNER + d] = p + uu * Dd;
    }
}

// ---------------- gate: y *= silu(res), res = xr[:, D_INNER:] ----------------
__global__ __launch_bounds__(256)
void gate_kernel(float* __restrict__ y, const float* __restrict__ xr, int total)
{
    int i = blockIdx.x * blockDim.x + threadIdx.x;
    if (i >= total) return;
    const int d = i % D_INNER;
    const int row = i / D_INNER;
    const float r = xr[(size_t)row * (2 * D_INNER) + D_INNER + d];
    y[i] *= silu_f(r);
}

// =====================================================================
static inline int gemm_lds_blocks(int M, int N) {
    return (M / GBM) * (N / GBN);
}
static inline int gemm_blocks(int M, int N, int NT) {
    int waves = (M / 16) * (N / (16 * NT));
    return (waves + 7) / 8;
}

extern "C" void kernel_launch(void* const* d_in, const int* in_sizes, int n_in,
                              void* d_out, int out_size, void* d_ws, size_t ws_size,
                              hipStream_t stream)
{
    const int*   text    = (const int*)  d_in[0];
    const float* visual  = (const float*)d_in[1];
    const float* emb     = (const float*)d_in[2];
    const float* norm_w  = (const float*)d_in[3];
    const float* norm_fw = (const float*)d_in[4];
    const float* in_proj = (const float*)d_in[5];
    const float* in_proj2= (const float*)d_in[6];
    const float* conv_w  = (const float*)d_in[7];
    const float* conv_b  = (const float*)d_in[8];
    const float* x_proj  = (const float*)d_in[9];
    const float* c_proj  = (const float*)d_in[10];
    const float* dt_w    = (const float*)d_in[11];
    const float* dt_b    = (const float*)d_in[12];
    const float* A_log   = (const float*)d_in[13];
    const float* Dp      = (const float*)d_in[14];
    const float* out_w   = (const float*)d_in[15];
    float* out = (float*)d_out;

    // ---- workspace layout (floats) ----
    float* ws = (float*)d_ws;
    size_t off = 0;
    float* x0   = ws + off; off += (size_t)ROWS * D_MODEL;
    float* x1   = ws + off; off += (size_t)ROWS * D_MODEL;
    float* q    = ws + off; off += (size_t)ROWS * D_MODEL;
    float* vbuf = ws + off; off += (size_t)ROWS * D_MODEL;
    float* xr   = ws + off; off += (size_t)ROWS * 2 * D_INNER;
    float* vin  = ws + off; off += (size_t)ROWS * D_INNER;
    float* tbuf = ws + off; off += (size_t)ROWS * D_INNER;
    float* xdbl = ws + off; off += (size_t)ROWS * (DT_RANK + D_STATE);
    float* cbuf = ws + off; off += (size_t)ROWS * D_STATE;
    float* dlt  = ws + off; off += (size_t)ROWS * D_INNER;
    float* ybuf = ws + off; off += (size_t)ROWS * D_INNER;
    (void)ws_size; (void)in_sizes; (void)n_in; (void)out_size;

    const int elemTot = ROWS * D_INNER;

    // 1) embedding gather
    {
        int total = ROWS * D_MODEL;
        embed_kernel<<<(total + 255) / 256, 256, 0, stream>>>(text, emb, x0, total);
    }

    float* xcur = x0;
    float* xnxt = x1;

    for (int i = 0; i < N_LAYER; ++i) {
        const float* nw   = norm_w   + (size_t)i * D_MODEL;
        const float* ipw  = in_proj  + (size_t)i * 2 * D_INNER * D_MODEL;
        const float* ipw2 = in_proj2 + (size_t)i * D_INNER * D_MODEL;
        const float* cw   = conv_w   + (size_t)i * D_INNER * D_CONV;
        const float* cb   = conv_b   + (size_t)i * D_INNER;
        const float* xpw  = x_proj   + (size_t)i * (DT_RANK + D_STATE) * D_INNER;
        const float* cpw  = c_proj   + (size_t)i * D_STATE * D_INNER;
        const float* dtw  = dt_w     + (size_t)i * D_INNER * DT_RANK;
        const float* dtb  = dt_b     + (size_t)i * D_INNER;
        const float* Al   = A_log    + (size_t)i * D_INNER * D_STATE;
        const float* Di   = Dp       + (size_t)i * D_INNER;
        const float* opw  = out_w    + (size_t)i * D_MODEL * D_INNER;

        rmsnorm_kernel<<<ROWS, 256, 0, stream>>>(xcur,   nw, q,    D_MODEL);
        rmsnorm_kernel<<<ROWS, 256, 0, stream>>>(visual, nw, vbuf, D_MODEL);

        // xr = q @ in_proj^T   (2048 x 3072, K=768)
        gemm_wmma_f32_lds<<<gemm_lds_blocks(ROWS, 2 * D_INNER), 256, 0, stream>>>(
            q, ipw, xr, nullptr, nullptr,
            ROWS, 2 * D_INNER, D_MODEL, D_MODEL, D_MODEL, 2 * D_INNER, 0);

        // vin = vbuf @ in_proj2^T (2048 x 1536, K=768)
        gemm_wmma_f32_lds<<<gemm_lds_blocks(ROWS, D_INNER), 256, 0, stream>>>(
            vbuf, ipw2, vin, nullptr, nullptr,
            ROWS, D_INNER, D_MODEL, D_MODEL, D_MODEL, D_INNER, 0);

        // t = silu(causal_conv(xr[:, :D_INNER]) + conv_b)
        conv_silu_kernel<<<(elemTot + 255) / 256, 256, 0, stream>>>(xr, cw, cb, tbuf, elemTot);

        // x_dbl = t @ x_proj^T (2048 x 64, K=1536)
        gemm_wmma_f32_lds<<<gemm_lds_blocks(ROWS, DT_RANK + D_STATE), 256, 0, stream>>>(
            tbuf, xpw, xdbl, nullptr, nullptr,
            ROWS, DT_RANK + D_STATE, D_INNER, D_INNER, D_INNER, DT_RANK + D_STATE, 0);

        // C = vin @ c_proj^T (2048 x 16, K=1536) -- small N, direct path
        gemm_wmma_f32<1><<<gemm_blocks(ROWS, D_STATE, 1), 256, 0, stream>>>(
            vin, cpw, cbuf, nullptr, nullptr,
            ROWS, D_STATE, D_INNER, D_INNER, D_INNER, D_STATE, 0);

        // delta = softplus(x_dbl[:, :48] @ dt_proj^T + dt_b)  (2048 x 1536, K=48)
        gemm_wmma_f32_lds<<<gemm_lds_blocks(ROWS, D_INNER), 256, 0, stream>>>(
            xdbl, dtw, dlt, dtb, nullptr,
            ROWS, D_INNER, DT_RANK, DT_RANK + D_STATE, DT_RANK, D_INNER, 2);

        // selective scan
        scan_kernel<<<BATCH * (D_INNER / 16), 256, 0, stream>>>(
            tbuf, dlt, xdbl, cbuf, Al, Di, ybuf);

        // y *= silu(res)
        gate_kernel<<<(elemTot + 255) / 256, 256, 0, stream>>>(ybuf, xr, elemTot);

        // x = y @ out_proj^T + x  (2048 x 768, K=1536)
        gemm_wmma_f32_lds<<<gemm_lds_blocks(ROWS, D_MODEL), 256, 0, stream>>>(
            ybuf, opw, xnxt, nullptr, xcur,
            ROWS, D_MODEL, D_INNER, D_INNER, D_INNER, D_MODEL, 0);

        float* tmp = xcur; xcur = xnxt; xnxt = tmp;
    }

    // final rmsnorm + tied LM head: out = rmsnorm(x) @ emb^T  (2048 x 32000, K=768)
    rmsnorm_kernel<<<ROWS, 256, 0, stream>>>(xcur, norm_fw, q, D_MODEL);
    gemm_wmma_f32_lds<<<gemm_lds_blocks(ROWS, VOCAB), 256, 0, stream>>>(
        q, emb, out, nullptr, nullptr,
        ROWS, VOCAB, D_MODEL, D_MODEL, D_MODEL, VOCAB, 0);
}